// MemoryAggregator_20160576487957
// MI455X (gfx1250) — compile-verified
//
#include <hip/hip_runtime.h>
#include <math.h>

// ---------------------------------------------------------------------------
// Graph attention aggregation for MI455X (gfx1250, wave32).
//   Phase 1: QKV projection [N,256]@[256,32]x3 via V_WMMA_F32_16X16X4_F32
//            (fp32 WMMA path keeps full precision for the exp() that follows)
//   Phase 2: per-edge logits + segment max (wave-per-edge, shfl reduction)
//   Phase 3: exp / segment-sum / weighted-V scatter (float atomics into L2)
//   Phase 4: safe divide (matches reference's empty-row -> zeros semantics)
// ---------------------------------------------------------------------------

typedef __attribute__((ext_vector_type(2))) float v2f;
typedef __attribute__((ext_vector_type(8))) float v8f;

#define D_IN 256
#define HS   32
#define XS_STRIDE 260          // 256 + 4 pad: lane stride 4 banks -> conflict-free b64 reads
#define INV_SQRT_HS 0.17677669529663687f

// ------------------------- Phase 1: QKV projection -------------------------
// One block = one 16-row tile of X, staged once into LDS.
// 6 waves per block = 6 output column tiles (Q cols 0-15/16-31, K, V).
// Each wave: 64 x V_WMMA_F32_16X16X4_F32 over K=256.
__global__ void __launch_bounds__(192)
qkv_proj_wmma(const float* __restrict__ X,
              const float* __restrict__ Wq,
              const float* __restrict__ Wk,
              const float* __restrict__ Wv,
              float* __restrict__ QKV, int N)
{
    __shared__ float xs[16 * XS_STRIDE];

    const int m0 = blockIdx.x * 16;

    // Cooperative, coalesced stage of X[m0:m0+16, :] into LDS.
    for (int f = threadIdx.x; f < 16 * D_IN; f += blockDim.x) {
        const int r = f >> 8;          // row in tile
        const int c = f & (D_IN - 1);  // column
        int row = m0 + r; if (row >= N) row = N - 1;   // clamp (N is 16-divisible anyway)
        xs[r * XS_STRIDE + c] = X[(size_t)row * D_IN + c];
    }
    __syncthreads();

    const int waveId = threadIdx.x >> 5;     // 0..5 -> column tile
    const int lane   = threadIdx.x & 31;
    const int half   = lane >> 4;            // K sub-pair select for A/B fragments
    const int lid    = lane & 15;

    const int mat  = waveId >> 1;                    // 0=Q, 1=K, 2=V
    const int ncol = (waveId & 1) * 16 + lid;        // column within [0,32)
    const float* __restrict__ W = (mat == 0) ? Wq : (mat == 1) ? Wk : Wv;

    v8f acc = {0.f, 0.f, 0.f, 0.f, 0.f, 0.f, 0.f, 0.f};

    #pragma unroll 8
    for (int k0 = 0; k0 < D_IN; k0 += 4) {
        const int ka = k0 + half * 2;                 // this half-wave's K pair
        // A fragment (16x4 f32): lane lid holds row M=lid, K={ka, ka+1} -> ds_load_b64
        const float* ap = &xs[lid * XS_STRIDE + ka];
        v2f a; a.x = ap[0]; a.y = ap[1];
        // B fragment (4x16 f32): lane lid holds col N=ncol, K={ka, ka+1}
        v2f b; b.x = W[ka * HS + ncol]; b.y = W[(ka + 1) * HS + ncol];
        acc = __builtin_amdgcn_wmma_f32_16x16x4_f32(
            /*neg_a=*/false, a, /*neg_b=*/false, b,
            /*c_mod=*/(short)0, acc, /*reuse_a=*/false, /*reuse_b=*/false);
    }

    // C/D layout: VGPR r -> row (r + half*8), lane lid -> column.
    // Uniform in-range check: avoids 8 per-row exec-mask toggles on the
    // hot path (always taken here since N % 16 == 0).
    float* __restrict__ out = QKV + (size_t)mat * N * HS;
    float* __restrict__ o   = out + (size_t)(m0 + half * 8) * HS + ncol;
    if (m0 + 16 <= N) {
        #pragma unroll
        for (int r = 0; r < 8; ++r) o[(size_t)r * HS] = acc[r];
    } else {
        #pragma unroll
        for (int r = 0; r < 8; ++r)
            if (m0 + half * 8 + r < N) o[(size_t)r * HS] = acc[r];
    }
}

// ------------------------- Phase 2/3 helpers -------------------------------
__device__ __forceinline__ void atomicMaxF(float* addr, float val) {
    // Monotonic int mapping: works with addr initialized to -inf.
    if (!(__float_as_int(val) < 0))
        atomicMax((int*)addr, __float_as_int(val));
    else
        atomicMin((unsigned int*)addr, (unsigned int)__float_as_int(val));
}

// One thread per 4 floats: b128 traffic. Covers rowmax/denom (N each) and
// the numerator accumulator in d_out (N*HS).
__global__ void init_state(float4* __restrict__ rowmax4,
                           float4* __restrict__ denom4,
                           float4* __restrict__ out4, int n4, int nOut4)
{
    const int idx = blockIdx.x * blockDim.x + threadIdx.x;
    if (idx < n4) {
        const float ninf = -__builtin_inff();
        rowmax4[idx] = make_float4(ninf, ninf, ninf, ninf);
        denom4[idx]  = make_float4(0.f, 0.f, 0.f, 0.f);
    }
    if (idx < nOut4) out4[idx] = make_float4(0.f, 0.f, 0.f, 0.f);
}

// One wave per edge: lane j owns head-dim j. Coalesced 128B row reads from
// L2-resident Q/K, butterfly reduction for the dot product.
__global__ void __launch_bounds__(256)
edge_scores_max(const float* __restrict__ Q, const float* __restrict__ K,
                const int* __restrict__ er, const int* __restrict__ ec,
                float* __restrict__ rowmax, int E)
{
    const int lane = threadIdx.x & 31;
    const int wave = (blockIdx.x * blockDim.x + threadIdx.x) >> 5;
    const int nWaves = (gridDim.x * blockDim.x) >> 5;
    for (int e = wave; e < E; e += nWaves) {
        const int r = er[e], c = ec[e];
        float p = Q[(size_t)r * HS + lane] * K[(size_t)c * HS + lane];
        #pragma unroll
        for (int off = 16; off > 0; off >>= 1) p += __shfl_xor(p, off, 32);
        if (lane == 0) atomicMaxF(&rowmax[r], p * INV_SQRT_HS);
    }
}

__global__ void __launch_bounds__(256)
edge_attend(const float* __restrict__ Q, const float* __restrict__ K,
            const float* __restrict__ V,
            const int* __restrict__ er, const int* __restrict__ ec,
            const float* __restrict__ rowmax,
            float* __restrict__ denom, float* __restrict__ num, int E)
{
    const int lane = threadIdx.x & 31;
    const int wave = (blockIdx.x * blockDim.x + threadIdx.x) >> 5;
    const int nWaves = (gridDim.x * blockDim.x) >> 5;
    for (int e = wave; e < E; e += nWaves) {
        const int r = er[e], c = ec[e];
        float p = Q[(size_t)r * HS + lane] * K[(size_t)c * HS + lane];
        #pragma unroll
        for (int off = 16; off > 0; off >>= 1) p += __shfl_xor(p, off, 32);
        const float s = p * INV_SQRT_HS;
        const float w = __expf(s - rowmax[r]);        // all lanes: same scalar
        atomicAdd(&num[(size_t)r * HS + lane], w * V[(size_t)c * HS + lane]);
        if (lane == 0) atomicAdd(&denom[r], w);
    }
}

// One thread per 4 output floats (HS=32 -> all 4 share one node's denom).
__global__ void finalize(const float* __restrict__ denom, float4* __restrict__ out4, int nOut4)
{
    const int idx = blockIdx.x * blockDim.x + threadIdx.x;
    if (idx < nOut4) {
        const float d = denom[idx >> 3];          // (idx*4) >> 5
        float4 v = out4[idx];
        if (d > 0.f) {
            const float inv = 1.f / fmaxf(d, 1e-30f);
            v.x *= inv; v.y *= inv; v.z *= inv; v.w *= inv;
        } else {
            v = make_float4(0.f, 0.f, 0.f, 0.f);
        }
        out4[idx] = v;
    }
}

// ---------------------------------------------------------------------------
extern "C" void kernel_launch(void* const* d_in, const int* in_sizes, int n_in,
                              void* d_out, int out_size, void* d_ws, size_t ws_size,
                              hipStream_t stream)
{
    const float* X  = (const float*)d_in[0];
    const float* Wq = (const float*)d_in[1];
    const float* Wk = (const float*)d_in[2];
    const float* Wv = (const float*)d_in[3];
    const int*   ei = (const int*)d_in[4];

    const int N = in_sizes[0] / D_IN;    // 50000
    const int E = in_sizes[4] / 2;       // 1600000
    const int* er = ei;                  // edge_index[0, :]
    const int* ec = ei + E;              // edge_index[1, :]

    // Workspace layout: Q | K | V | rowmax | denom  (~19.6 MB)
    float* ws     = (float*)d_ws;
    float* Q      = ws;
    float* K      = Q + (size_t)N * HS;
    float* V      = K + (size_t)N * HS;
    float* rowmax = V + (size_t)N * HS;
    float* denom  = rowmax + N;
    float* out    = (float*)d_out;       // doubles as the numerator accumulator

    // Phase 1: QKV projection (WMMA f32). One block per 16-row tile.
    const int rowTiles = (N + 15) / 16;
    qkv_proj_wmma<<<rowTiles, 192, 0, stream>>>(X, Wq, Wk, Wv, Q, N);

    // Phase 2 init: rowmax=-inf, denom=0, out=0 (vectorized; N*HS % 4 == 0).
    const int n4    = N / 4;             // N % 4 == 0 for this problem
    const int nOut4 = (N * HS) / 4;
    init_state<<<(nOut4 + 255) / 256, 256, 0, stream>>>(
        (float4*)rowmax, (float4*)denom, (float4*)out, n4, nOut4);

    // Phase 2: logits + segment max. One wave per edge (8 waves/block).
    const int edgeBlocks = (E + 7) / 8;
    edge_scores_max<<<edgeBlocks, 256, 0, stream>>>(Q, K, er, ec, rowmax, E);

    // Phase 3: exp + denom + weighted-V scatter.
    edge_attend<<<edgeBlocks, 256, 0, stream>>>(Q, K, V, er, ec, rowmax, denom, out, E);

    // Phase 4: safe divide.
    finalize<<<(nOut4 + 255) / 256, 256, 0, stream>>>(denom, (float4*)out, nOut4);
}